// IWT_33217277067815
// MI455X (gfx1250) — compile-verified
//
#include <hip/hip_runtime.h>
#include <stdint.h>

#ifndef __has_builtin
#define __has_builtin(x) 0
#endif

typedef float v4f __attribute__((ext_vector_type(4)));
typedef int   v4i __attribute__((ext_vector_type(4)));

// Use the gfx1250 async global->LDS pipe when the toolchain exposes it.
#if defined(__HIP_DEVICE_COMPILE__) && \
    __has_builtin(__builtin_amdgcn_global_load_async_to_lds_b128)
#define IWT_ASYNC_LDS 1
#else
#define IWT_ASYNC_LDS 0
#endif

// Problem constants (fixed by the reference): B=16, C4=256 (c=64), H=W=128.
#define IWT_B   16
#define IWT_C   64
#define IWT_H   128
#define IWT_W   128

__global__ __launch_bounds__(256) void IWT_kernel(const float* __restrict__ x,
                                                  float* __restrict__ out) {
    constexpr int PLANE  = IWT_H * IWT_W;       // 16384 floats per channel plane
    constexpr int GROUP  = IWT_C * PLANE;       // offset between a/b/c/d groups
    constexpr int OW     = 2 * IWT_W;           // 256
    constexpr int OUT_CH = (2 * IWT_H) * OW;    // 65536 floats per output channel

    const int g  = blockIdx.x * 256 + threadIdx.x;
    const int wq = g & 31;            // quad index along W (W/4 = 32)
    const int h  = (g >> 5)  & (IWT_H - 1);
    const int ch = (g >> 12) & (IWT_C - 1);
    const int b  = g >> 18;

    const size_t in_off = (size_t)b * (size_t)(4 * GROUP)
                        + (size_t)ch * PLANE
                        + (size_t)h * IWT_W
                        + (size_t)wq * 4;
    const float* gsrc = x + in_off;

    v4f a, bb, cc, dd;

#if IWT_ASYNC_LDS
    // Stage the four 16B quads through LDS via the async global->LDS pipe.
    // Plane-major slots: lane t reads back at byte 16*t -> conflict-free b128.
    __shared__ v4f lds4[4 * 256];
    const unsigned tid = threadIdx.x;

    __builtin_amdgcn_global_load_async_to_lds_b128(
        (__attribute__((address_space(1))) v4i*)(gsrc),
        (__attribute__((address_space(3))) v4i*)(&lds4[0 * 256 + tid]), 0, 0);
    __builtin_amdgcn_global_load_async_to_lds_b128(
        (__attribute__((address_space(1))) v4i*)(gsrc + GROUP),
        (__attribute__((address_space(3))) v4i*)(&lds4[1 * 256 + tid]), 0, 0);
    __builtin_amdgcn_global_load_async_to_lds_b128(
        (__attribute__((address_space(1))) v4i*)(gsrc + 2 * GROUP),
        (__attribute__((address_space(3))) v4i*)(&lds4[2 * 256 + tid]), 0, 0);
    __builtin_amdgcn_global_load_async_to_lds_b128(
        (__attribute__((address_space(1))) v4i*)(gsrc + 3 * GROUP),
        (__attribute__((address_space(3))) v4i*)(&lds4[3 * 256 + tid]), 0, 0);

#if __has_builtin(__builtin_amdgcn_s_wait_asynccnt)
    __builtin_amdgcn_s_wait_asynccnt(0);
#else
    asm volatile("s_wait_asynccnt 0" ::: "memory");
#endif
    asm volatile("" ::: "memory");  // keep LDS reads after the wait

    a  = lds4[0 * 256 + tid];
    bb = lds4[1 * 256 + tid];
    cc = lds4[2 * 256 + tid];
    dd = lds4[3 * 256 + tid];
#else
    // Direct streaming fallback: coalesced 128-bit non-temporal loads.
    a  = __builtin_nontemporal_load((const v4f*)(gsrc));
    bb = __builtin_nontemporal_load((const v4f*)(gsrc + GROUP));
    cc = __builtin_nontemporal_load((const v4f*)(gsrc + 2 * GROUP));
    dd = __builtin_nontemporal_load((const v4f*)(gsrc + 3 * GROUP));
#endif

    // Butterfly: 8 adds + 4 muls per quad (vs 12 naive adds).
    const v4f u0 = a + bb;   // a+b
    const v4f u1 = a - bb;   // a-b
    const v4f w0 = cc + dd;  // c+d
    const v4f w1 = cc - dd;  // c-d
    const v4f h00 = 0.5f * (u1 - w1);  // a-b-c+d
    const v4f h01 = 0.5f * (u0 - w0);  // a+b-c-d
    const v4f h10 = 0.5f * (u1 + w1);  // a-b+c-d
    const v4f h11 = 0.5f * (u0 + w0);  // a+b+c+d

    // Interleave in-register: even row = {h00,h01,...}, odd row = {h10,h11,...}.
    const v4f e0 = {h00.x, h01.x, h00.y, h01.y};
    const v4f e1 = {h00.z, h01.z, h00.w, h01.w};
    const v4f o0 = {h10.x, h11.x, h10.y, h11.y};
    const v4f o1 = {h10.z, h11.z, h10.w, h11.w};

    const size_t ob = (size_t)b * (size_t)(IWT_C * OUT_CH)
                    + (size_t)ch * OUT_CH
                    + (size_t)(2 * h) * OW
                    + (size_t)wq * 8;

    // Streamed-once output: non-temporal b128 stores (don't thrash 192MB L2).
    __builtin_nontemporal_store(e0, (v4f*)(out + ob));
    __builtin_nontemporal_store(e1, (v4f*)(out + ob + 4));
    __builtin_nontemporal_store(o0, (v4f*)(out + ob + OW));
    __builtin_nontemporal_store(o1, (v4f*)(out + ob + OW + 4));
}

extern "C" void kernel_launch(void* const* d_in, const int* in_sizes, int n_in,
                              void* d_out, int out_size, void* d_ws, size_t ws_size,
                              hipStream_t stream) {
    (void)in_sizes; (void)n_in; (void)d_ws; (void)ws_size; (void)out_size;
    const float* x = (const float*)d_in[0];
    float* out = (float*)d_out;

    // Total 4-wide quads: 16 * 64 * 128 * 32 = 2^22.
    constexpr int NQ = IWT_B * IWT_C * IWT_H * (IWT_W / 4);
    dim3 grid(NQ / 256), block(256);
    IWT_kernel<<<grid, block, 0, stream>>>(x, out);
}